// GCNNet_78245714198553
// MI455X (gfx1250) — compile-verified
//
#include <hip/hip_runtime.h>
#include <math.h>

typedef float v2f __attribute__((ext_vector_type(2)));
typedef float v8f __attribute__((ext_vector_type(8)));

#define IN_FEATS  256
#define OUT_FEATS 64
#define WPAIRS    (IN_FEATS / 2)   // 128 K-pairs
#define WSTRIDE   160              // floats per pair-row (padded: bank-conflict-free)

// ---------------------------------------------------------------------------
// Kernel 1: edge -> degree accumulation (segment_sum of ones)
// ---------------------------------------------------------------------------
__global__ void gcn_degrees(const int* __restrict__ src,
                            const int* __restrict__ dst,
                            float* __restrict__ deg_src,
                            float* __restrict__ deg_dst,
                            int n_edges) {
    int t = blockIdx.x * blockDim.x + threadIdx.x;
    if (t >= n_edges) return;
    atomicAdd(&deg_src[src[t]], 1.0f);
    atomicAdd(&deg_dst[dst[t]], 1.0f);
}

// ---------------------------------------------------------------------------
// Kernel 2: deg -> rsqrt(max(deg,1)) in place, both arrays
// ---------------------------------------------------------------------------
__global__ void gcn_norms(float* __restrict__ deg_src,
                          float* __restrict__ deg_dst,
                          int n_nodes) {
    int t = blockIdx.x * blockDim.x + threadIdx.x;
    if (t >= n_nodes) return;
    deg_src[t] = rsqrtf(fmaxf(deg_src[t], 1.0f));
    deg_dst[t] = rsqrtf(fmaxf(deg_dst[t], 1.0f));
}

// ---------------------------------------------------------------------------
// Kernel 3: h = (feat * norm_src[:,None]) @ W   via V_WMMA_F32_16X16X4_F32
//   one wave32 per 16-row tile; 4 N-tiles of 16 (OUT_FEATS=64); K in steps of 4
//   A 16x4 f32 layout: lanes 0-15 = {K=k,k+1}, lanes 16-31 = {K=k+2,k+3}
//   B fragments assembled from K-pair-interleaved LDS image of W:
//     Wl[p*WSTRIDE + 2n + 0] = W[2p][n], Wl[p*WSTRIDE + 2n + 1] = W[2p+1][n]
//   -> each B operand is one aligned ds_load_b64, zero register shuffles.
//   C/D: VGPR r -> M = r + 8*(lane>=16), N = lane%16
// ---------------------------------------------------------------------------
__global__ void __launch_bounds__(256)
gcn_gemm_wmma(const float* __restrict__ feat,
              const float* __restrict__ norm_src,
              const float* __restrict__ W,
              float* __restrict__ h,
              int n_tiles) {
    __shared__ float Wl[WPAIRS * WSTRIDE];   // 80 KB of 320 KB WGP LDS

    const int tid = threadIdx.x;
    // Stage W K-pair interleaved: 128 pairs x 32 n-duos = 4096 quads, 16/thread
    for (int i = tid; i < WPAIRS * (OUT_FEATS / 2); i += 256) {
        int p  = i >> 5;              // K-pair 0..127
        int n2 = (i & 31) * 2;        // even column 0..62
        float2 lo = *reinterpret_cast<const float2*>(&W[(2 * p) * OUT_FEATS + n2]);
        float2 hi = *reinterpret_cast<const float2*>(&W[(2 * p + 1) * OUT_FEATS + n2]);
        float4 q;  q.x = lo.x;  q.y = hi.x;  q.z = lo.y;  q.w = hi.y;
        *reinterpret_cast<float4*>(&Wl[p * WSTRIDE + 2 * n2]) = q;
    }
    __syncthreads();

    const int wave = tid >> 5;
    const int lane = tid & 31;
    const int tile = blockIdx.x * 8 + wave;
    if (tile >= n_tiles) return;      // wave-uniform exit; WMMA waves keep EXEC=~0

    const int half = lane >> 4;       // 0: lanes 0-15 (K=k,k+1), 1: lanes 16-31 (K=k+2,k+3)
    const int l15  = lane & 15;
    const int m    = tile * 16 + l15; // row this lane owns for A
    const float nrm = norm_src[m];
    const float* frow  = feat + (size_t)m * IN_FEATS + 2 * half;
    const float* wbase = &Wl[half * WSTRIDE + 2 * l15];  // pair-row p = k/2 + half

    v8f c0 = {}, c1 = {}, c2 = {}, c3 = {};

    #pragma unroll 4
    for (int k = 0; k < IN_FEATS; k += 4) {
        // A fragment (fold in src normalization)
        v2f a = *reinterpret_cast<const v2f*>(frow + k);
        a = a * nrm;

        const float* wp = wbase + (k >> 1) * WSTRIDE;
        v2f b0 = *reinterpret_cast<const v2f*>(wp);        // N-tile 0
        v2f b1 = *reinterpret_cast<const v2f*>(wp + 32);   // N-tile 1
        v2f b2 = *reinterpret_cast<const v2f*>(wp + 64);   // N-tile 2
        v2f b3 = *reinterpret_cast<const v2f*>(wp + 96);   // N-tile 3

        c0 = __builtin_amdgcn_wmma_f32_16x16x4_f32(false, a, false, b0,
                                                   (short)0, c0, false, false);
        c1 = __builtin_amdgcn_wmma_f32_16x16x4_f32(false, a, false, b1,
                                                   (short)0, c1, false, false);
        c2 = __builtin_amdgcn_wmma_f32_16x16x4_f32(false, a, false, b2,
                                                   (short)0, c2, false, false);
        c3 = __builtin_amdgcn_wmma_f32_16x16x4_f32(false, a, false, b3,
                                                   (short)0, c3, false, false);
    }

    // store D: row = tile*16 + r + 8*half, col = j*16 + l15
    const int rbase = tile * 16 + 8 * half;
    #pragma unroll
    for (int r = 0; r < 8; ++r) {
        float* orow = h + (size_t)(rbase + r) * OUT_FEATS + l15;
        orow[0]  = c0[r];
        orow[16] = c1[r];
        orow[32] = c2[r];
        orow[48] = c3[r];
    }
}

// ---------------------------------------------------------------------------
// Kernel 4: SpMM scatter  agg[dst] += h[src]  (16 lanes per edge, float4 each)
//   h (12.8 MB) fits in the 192 MB L2 -> gather + f32 atomics run at L2 rate
// ---------------------------------------------------------------------------
__global__ void gcn_scatter(const int* __restrict__ src,
                            const int* __restrict__ dst,
                            const float* __restrict__ h,
                            float* __restrict__ agg,
                            int n_edges) {
    long long t = (long long)blockIdx.x * blockDim.x + threadIdx.x;
    int e = (int)(t >> 4);
    if (e >= n_edges) return;
    int q = (int)(t & 15);
    int s = src[e];
    int d = dst[e];
    float4 v = *reinterpret_cast<const float4*>(h + (size_t)s * OUT_FEATS + q * 4);
    float* ap = agg + (size_t)d * OUT_FEATS + q * 4;
    atomicAdd(ap + 0, v.x);
    atomicAdd(ap + 1, v.y);
    atomicAdd(ap + 2, v.z);
    atomicAdd(ap + 3, v.w);
}

// ---------------------------------------------------------------------------
// Kernel 5: out = sigmoid(agg * norm_dst[:,None] + b)
// ---------------------------------------------------------------------------
__global__ void gcn_finalize(const float* __restrict__ agg,
                             const float* __restrict__ norm_dst,
                             const float* __restrict__ b,
                             float* __restrict__ out,
                             int total) {
    int t = blockIdx.x * blockDim.x + threadIdx.x;
    if (t >= total) return;
    int node = t >> 6;       // /OUT_FEATS
    int f    = t & 63;
    float x = agg[t] * norm_dst[node] + b[f];
    out[t] = 1.0f / (1.0f + __expf(-x));
}

// ---------------------------------------------------------------------------
// Launch
// ---------------------------------------------------------------------------
extern "C" void kernel_launch(void* const* d_in, const int* in_sizes, int n_in,
                              void* d_out, int out_size, void* d_ws, size_t ws_size,
                              hipStream_t stream) {
    const float* feat = (const float*)d_in[0];
    const int*   src  = (const int*)d_in[1];
    const int*   dst  = (const int*)d_in[2];
    const float* W    = (const float*)d_in[3];
    const float* bias = (const float*)d_in[4];
    float*       out  = (float*)d_out;

    const int n_nodes = in_sizes[0] / IN_FEATS;   // 50000
    const int n_edges = in_sizes[1];              // 800000

    // workspace layout: [norm_src n][norm_dst n][agg n*64]
    float* norm_src = (float*)d_ws;
    float* norm_dst = norm_src + n_nodes;
    float* agg      = norm_dst + n_nodes;

    size_t zero_bytes = ((size_t)2 * n_nodes + (size_t)n_nodes * OUT_FEATS)
                        * sizeof(float);
    hipMemsetAsync(d_ws, 0, zero_bytes, stream);

    // 1) degrees
    gcn_degrees<<<(n_edges + 255) / 256, 256, 0, stream>>>(
        src, dst, norm_src, norm_dst, n_edges);

    // 2) norms
    gcn_norms<<<(n_nodes + 255) / 256, 256, 0, stream>>>(
        norm_src, norm_dst, n_nodes);

    // 3) GEMM: h staged in d_out (same shape as output)
    const int n_tiles = n_nodes / 16;             // 3125, exact
    gcn_gemm_wmma<<<(n_tiles + 7) / 8, 256, 0, stream>>>(
        feat, norm_src, W, out, n_tiles);

    // 4) edge scatter-add into agg
    long long scat_threads = (long long)n_edges * 16;
    gcn_scatter<<<(int)((scat_threads + 255) / 256), 256, 0, stream>>>(
        src, dst, out, agg, n_edges);

    // 5) finalize (overwrites d_out after it has been consumed)
    int total = n_nodes * OUT_FEATS;
    gcn_finalize<<<(total + 255) / 256, 256, 0, stream>>>(
        agg, norm_dst, bias, out, total);
}